// SingleGCN_17712445129197
// MI455X (gfx1250) — compile-verified
//
#include <hip/hip_runtime.h>

#define N_NODES 50000
#define D 128          // D_IN == D_OUT == S
#define OUT_W 256      // S + D_OUT

typedef float v2f __attribute__((ext_vector_type(2)));
typedef float v8f __attribute__((ext_vector_type(8)));

// ---------------------------------------------------------------------------
// Kernel 1: normalized = hidden / degree ; agg = hidden (residual seed) ;
//           out[:, 0:128] = stack.  One float4 per thread.
// ---------------------------------------------------------------------------
__global__ void __launch_bounds__(256)
gcn_prep(const float* __restrict__ hidden,
         const float* __restrict__ degree,   // [N,1]
         const float* __restrict__ stack,
         float* __restrict__ normalized,
         float* __restrict__ agg,
         float* __restrict__ out)
{
    int t = blockIdx.x * blockDim.x + threadIdx.x;
    const int total = N_NODES * (D / 4);
    if (t >= total) return;
    int n  = t / (D / 4);
    int c4 = (t % (D / 4)) * 4;

    const float4 h = *(const float4*)(hidden + (size_t)n * D + c4);
    const float4 s = *(const float4*)(stack  + (size_t)n * D + c4);
    float dg = degree[n];

    float4 nm = make_float4(h.x / dg, h.y / dg, h.z / dg, h.w / dg);
    *(float4*)(normalized + (size_t)n * D + c4) = nm;
    *(float4*)(agg        + (size_t)n * D + c4) = h;     // residual term
    *(float4*)(out        + (size_t)n * OUT_W + c4) = s; // stack half
}

// ---------------------------------------------------------------------------
// Kernel 2: edge scatter.  One wave32 per edge; lane owns 4 contiguous
// features -> coalesced 512B gather of normalized[src], 128 contiguous
// f32 atomic adds into agg[dst] (L2-resident).
// ---------------------------------------------------------------------------
__global__ void __launch_bounds__(256)
gcn_scatter(const int* __restrict__ src,
            const int* __restrict__ dst,
            const float* __restrict__ normalized,
            float* __restrict__ agg,
            int E)
{
    int t    = blockIdx.x * blockDim.x + threadIdx.x;
    int e    = t >> 5;
    int lane = t & 31;
    if (e >= E) return;

    int s = src[e];
    int d = dst[e];

    const float4 m = *(const float4*)(normalized + (size_t)s * D + lane * 4);
    float* ap = agg + (size_t)d * D + lane * 4;
    atomicAdd(ap + 0, m.x);
    atomicAdd(ap + 1, m.y);
    atomicAdd(ap + 2, m.z);
    atomicAdd(ap + 3, m.w);
}

// ---------------------------------------------------------------------------
// Kernel 3: out[:, 128:256] = agg @ W + b using V_WMMA_F32_16X16X4_F32.
// One wave per 16x16 tile; block = 8 waves = one 16-row stripe, all 128 cols.
// A (16x4 f32): lane = M (mod 16), v0/v1 = K = 2*half .. 2*half+1
// B ( 4x16 f32): lane = N (mod 16), v0/v1 = K = 2*half .. 2*half+1
// C/D (16x16 f32): c[r] = element (M = r + 8*half, N = lane mod 16)
// ---------------------------------------------------------------------------
__global__ void __launch_bounds__(256)
gcn_gemm(const float* __restrict__ agg,   // reduced, [N,128]
         const float* __restrict__ Wm,    // [128,128] row-major
         const float* __restrict__ bias,  // [128]
         float* __restrict__ out)         // [N,256]
{
    int wave = threadIdx.x >> 5;          // 0..7 -> column tile
    int lane = threadIdx.x & 31;
    int row0 = blockIdx.x * 16;
    int col0 = wave * 16;

    int m    = lane & 15;
    int half = lane >> 4;                 // 0 or 1

    v8f c = {0.f, 0.f, 0.f, 0.f, 0.f, 0.f, 0.f, 0.f};

    const float* arow = agg + (size_t)(row0 + m) * D;

    #pragma unroll
    for (int k = 0; k < D; k += 4) {
        v2f a = *(const v2f*)(arow + k + 2 * half);
        v2f b;
        b.x = Wm[(size_t)(k + 2 * half)     * D + col0 + m];
        b.y = Wm[(size_t)(k + 2 * half + 1) * D + col0 + m];
        c = __builtin_amdgcn_wmma_f32_16x16x4_f32(
                /*neg_a=*/false, a, /*neg_b=*/false, b,
                /*c_mod=*/(short)0, c, /*reuse_a=*/false, /*reuse_b=*/false);
    }

    float bv  = bias[col0 + m];
    int   col = col0 + m;
    #pragma unroll
    for (int r = 0; r < 8; ++r) {
        int row = row0 + r + 8 * half;
        out[(size_t)row * OUT_W + D + col] = c[r] + bv;
    }
}

// ---------------------------------------------------------------------------
extern "C" void kernel_launch(void* const* d_in, const int* in_sizes, int n_in,
                              void* d_out, int out_size, void* d_ws, size_t ws_size,
                              hipStream_t stream)
{
    const float* hidden = (const float*)d_in[0];
    const float* degree = (const float*)d_in[1];
    const float* stack  = (const float*)d_in[2];
    const float* Wm     = (const float*)d_in[3];
    const float* bias   = (const float*)d_in[4];
    const int*   src    = (const int*)d_in[5];
    const int*   dst    = (const int*)d_in[6];
    float*       out    = (float*)d_out;
    const int    E      = in_sizes[5];

    float* normalized = (float*)d_ws;                     // 25.6 MB
    float* agg        = normalized + (size_t)N_NODES * D; // 25.6 MB

    // 1) normalize + residual seed + stack copy
    {
        int threads = N_NODES * (D / 4);
        gcn_prep<<<(threads + 255) / 256, 256, 0, stream>>>(
            hidden, degree, stack, normalized, agg, out);
    }
    // 2) edge scatter (wave per edge)
    {
        long long threads = (long long)E * 32;
        gcn_scatter<<<(unsigned)((threads + 255) / 256), 256, 0, stream>>>(
            src, dst, normalized, agg, E);
    }
    // 3) WMMA f32 GEMM + bias into out[:, 128:256]
    {
        gcn_gemm<<<N_NODES / 16, 256, 0, stream>>>(agg, Wm, bias, out);
    }
}